// SoftDTWSemanticAttention_54640573939740
// MI455X (gfx1250) — compile-verified
//
#include <hip/hip_runtime.h>
#include <math.h>

typedef __attribute__((ext_vector_type(2))) float v2f;
typedef __attribute__((ext_vector_type(4))) float v4f;
typedef __attribute__((ext_vector_type(8))) float v8f;

#define B_ 4
#define T_ 128
#define N_ 48
#define D_ 64
#define P_ 1176          // 48*49/2
#define WIN 4
#define GAMMA_F 0.5f
#define TOPL 5

// ---------------------------------------------------------------- kernel 1
__global__ void zero_f4_kernel(float* __restrict__ out, int n4) {
    int idx = blockIdx.x * blockDim.x + threadIdx.x;
    int stride = gridDim.x * blockDim.x;
    v4f z = {0.0f, 0.0f, 0.0f, 0.0f};
    for (int i = idx; i < n4; i += stride)
        ((v4f*)out)[i] = z;
}

// ---------------------------------------------------------------- kernel 2
// One wave32 per (batch, triu-pair). Computes banded soft-DTW distance and
// scatters it into the symmetric C matrix in workspace.
__global__ __launch_bounds__(32) void pair_sdtw_kernel(
        const float* __restrict__ X,   // X_sp [B,T,N,D]
        float* __restrict__ C)         // [B,N,N] workspace
{
    __shared__ float band[T_ * 9];     // band[t*9+k] = <A_t, B_{t-4+k}>
    __shared__ float an[T_];
    __shared__ float bn[T_];
    __shared__ float prev[T_ + 1];

    const int blk = blockIdx.x;
    const int b   = blk / P_;
    int p = blk % P_;

    // decode triu pair (ii <= jj), row-major over rows of length N-ii
    int ii = 0;
    while (p >= (N_ - ii)) { p -= (N_ - ii); ++ii; }
    const int jj = ii + p;

    const int lane = threadIdx.x;
    const int lm   = lane & 15;
    const int half = lane >> 4;

    const size_t rs = (size_t)N_ * D_;                 // stride between t rows
    const float* Ab = X + (size_t)b * T_ * rs + (size_t)ii * D_;
    const float* Bb = X + (size_t)b * T_ * rs + (size_t)jj * D_;

    // ---- row norms (VALU) ----
    for (int r = lane; r < T_; r += 32) {
        const float* ar = Ab + (size_t)r * rs;
        const float* br = Bb + (size_t)r * rs;
        float sa = 0.0f, sb = 0.0f;
#pragma unroll
        for (int d = 0; d < D_; d += 4) {
            v4f a = *(const v4f*)(ar + d);
            v4f q = *(const v4f*)(br + d);
            sa += a[0]*a[0] + a[1]*a[1] + a[2]*a[2] + a[3]*a[3];
            sb += q[0]*q[0] + q[1]*q[1] + q[2]*q[2] + q[3]*q[3];
        }
        an[r] = sa;
        bn[r] = sb;
    }
    __syncthreads();

    // ---- banded Gram via V_WMMA_F32_16X16X4_F32 ----
    // For each 16-row tile ti, only tiles tj in {ti-1, ti, ti+1} intersect
    // the |t-s|<=4 band.
    for (int ti = 0; ti < 8; ++ti) {
        v2f afr[16];
        const float* arow = Ab + (size_t)(16 * ti + lm) * rs;
#pragma unroll
        for (int kk = 0; kk < 16; ++kk)
            afr[kk] = *(const v2f*)(arow + kk * 4 + 2 * half);

        const int tj0 = (ti > 0) ? ti - 1 : 0;
        const int tj1 = (ti < 7) ? ti + 1 : 7;
        for (int tj = tj0; tj <= tj1; ++tj) {
            const float* brow = Bb + (size_t)(16 * tj + lm) * rs;
            v8f acc = {};
#pragma unroll
            for (int kk = 0; kk < 16; ++kk) {
                v2f bfr = *(const v2f*)(brow + kk * 4 + 2 * half);
                acc = __builtin_amdgcn_wmma_f32_16x16x4_f32(
                        false, afr[kk], false, bfr, (short)0, acc, false, false);
            }
            // scatter band entries: lane holds D[m= r+8*half][n= lm] of tile
#pragma unroll
            for (int r = 0; r < 8; ++r) {
                int t  = 16 * ti + r + 8 * half;
                int s  = 16 * tj + lm;
                int dl = s - t;
                if (dl >= -WIN && dl <= WIN)
                    band[t * 9 + dl + WIN] = acc[r];
            }
        }
    }
    __syncthreads();

    // ---- banded soft-DTW DP ----
    for (int j = lane; j <= T_; j += 32)
        prev[j] = (j == 0) ? 0.0f : __builtin_inff();
    __syncthreads();

    for (int i = 1; i <= T_; ++i) {
        const int  j     = i - WIN + lane;
        const bool valid = (lane < 9) && (j >= 1) && (j <= T_);
        float val = 0.0f;
        if (valid) {
            float r0 = prev[j - 1];      // diagonal predecessor
            float r2 = prev[j];          // vertical predecessor
            float m  = fminf(r0, r2);
            float sm;
            if (m > 1.0e37f) {
                sm = __builtin_inff();
            } else {
                // softmin = m - g*log(exp(-(r0-m)/g) + exp(-(r2-m)/g)), g=0.5
                sm = m - GAMMA_F * logf(expf(-(r0 - m) * 2.0f) +
                                        expf(-(r2 - m) * 2.0f));
            }
            float d2 = an[i - 1] + bn[j - 1] - 2.0f * band[(i - 1) * 9 + lane];
            val = sqrtf(fmaxf(d2, 0.0f)) + sm;
        }
        // same-wave LDS ops are in-order: all reads above precede these writes
        if (valid) prev[j] = val;
        if (i == 1 && lane == 0) prev[0] = __builtin_inff();  // row prepends inf
    }
    __syncthreads();

    if (lane == 0) {
        float res = prev[T_];
        C[((size_t)b * N_ + ii) * N_ + jj] = res;
        C[((size_t)b * N_ + jj) * N_ + ii] = res;
    }
}

// ---------------------------------------------------------------- kernel 3
// One thread per (b, n): softmax(-C), top-5 smallest of C+eye*1e9,
// mask + renormalize.
__global__ void finalize_kernel(const float* __restrict__ C,
                                float* __restrict__ outA) {
    const int b = blockIdx.x;
    const int n = threadIdx.x;
    if (n >= N_) return;

    const float* row = C + ((size_t)b * N_ + n) * N_;
    float c[N_];
    float mn = 3.4e38f;
    for (int j = 0; j < N_; ++j) { c[j] = row[j]; mn = fminf(mn, c[j]); }

    float e[N_];
    float sum = 0.0f;
    for (int j = 0; j < N_; ++j) { e[j] = expf(mn - c[j]); sum += e[j]; }

    bool used[N_];
    for (int j = 0; j < N_; ++j) used[j] = false;
    for (int r = 0; r < TOPL; ++r) {
        float best = 3.4e38f;
        int   bi   = 0;
        for (int j = 0; j < N_; ++j) {
            if (used[j]) continue;
            float cd = c[j] + ((j == n) ? 1.0e9f : 0.0f);
            if (cd < best) { best = cd; bi = j; }   // ties -> lowest index
        }
        used[bi] = true;
    }

    float msum = 0.0f;
    for (int j = 0; j < N_; ++j)
        if (used[j]) msum += e[j] / sum;
    float inv = 1.0f / (msum + 1e-12f);

    float* orow = outA + ((size_t)b * N_ + n) * N_;
    for (int j = 0; j < N_; ++j)
        orow[j] = used[j] ? (e[j] / sum) * inv : 0.0f;
}

// ---------------------------------------------------------------- launcher
extern "C" void kernel_launch(void* const* d_in, const int* in_sizes, int n_in,
                              void* d_out, int out_size, void* d_ws, size_t ws_size,
                              hipStream_t stream) {
    (void)in_sizes; (void)n_in; (void)out_size; (void)ws_size;
    // inputs: d_in[0] = H_temp (unused; H_sem is zeros), d_in[1] = X_sp
    const float* X_sp = (const float*)d_in[1];
    float* out = (float*)d_out;
    float* Cws = (float*)d_ws;                       // B*N*N floats = 36 KB

    const int HTOT = B_ * T_ * N_ * D_;              // 1,572,864 floats

    // 1) H_sem = zeros
    zero_f4_kernel<<<512, 256, 0, stream>>>(out, HTOT / 4);
    // 2) pairwise banded soft-DTW costs -> C
    pair_sdtw_kernel<<<B_ * P_, 32, 0, stream>>>(X_sp, Cws);
    // 3) softmax / top-k mask / renormalize -> A_sem
    finalize_kernel<<<B_, 64, 0, stream>>>(Cws, out + HTOT);
}